// GGNN_20873541058991
// MI455X (gfx1250) — compile-verified
//
#include <hip/hip_runtime.h>
#include <cstddef>

typedef float v2f __attribute__((ext_vector_type(2)));
typedef float v4f __attribute__((ext_vector_type(4)));
typedef float v8f __attribute__((ext_vector_type(8)));

#define NT    128   // 4 waves of 32
#define WAVES 4

// shared layout offsets (floats). Weight matrices stored pair-interleaved:
//   P[q][col][j] = WT[2q+j][col]   (q = k/2, j = k&1)  -> B frag = one aligned b64
#define OFF_W3UT 0        // 16 pairs x 32 cols x 2 = 1024
#define OFF_W5UT 1024
#define OFF_W3WT 2048     // 32 pairs x 32 x 2 = 2048
#define OFF_W4WT 4096
#define OFF_W5WT 6144
#define OFF_WOT  8192     // 32 pairs x 16 x 2 = 1024
#define OFF_WRXT 9216     // 16 pairs x 16 x 2 = 512
#define OFF_WRYT 9728
#define OFF_MAT  10240    // 16x80 row-major (A operand only)
#define OFF_MATT 11520    // 80x16 row-major (A operand only)
#define OFF_BIAS 12800    // b3w,b3u,b4w,b5w,b5u (32 each), bo,brx,bry,wr2 (16 each), br2
#define OFF_WAVE 13056
#define WAVE_FL  6656     // nodes 3072 + aBuf 3072 + rnB 512
#define SMEM_FL  (OFF_WAVE + WAVES * WAVE_FL)
#define SMEM_BYTES (SMEM_FL * 4)

__device__ __forceinline__ v8f wmma4(v2f a, v2f b, v8f c) {
  // D = A(16x4,f32) * B(4x16,f32) + C(16x16,f32)
  return __builtin_amdgcn_wmma_f32_16x16x4_f32(false, a, false, b, (short)0, c,
                                               false, false);
}
// A fragment (16x4): lanes 0-15 rows, VGPR0=K0(lo)/K2(hi), VGPR1=K1/K3 -> contiguous b64
__device__ __forceinline__ v2f ldsA(const float* base, int ld, int m0, int k0, int lane) {
  return *(const v2f*)(base + (m0 + (lane & 15)) * ld + k0 + ((lane >> 4) << 1));
}
__device__ __forceinline__ v2f glbA(const float* base, int ld, int m0, int k0, int lane) {
  return *(const v2f*)(base + (m0 + (lane & 15)) * ld + k0 + ((lane >> 4) << 1));
}
// B fragment from pair-interleaved weights: one aligned b64 per lane
__device__ __forceinline__ v2f ldsBP(const float* base, int ncol2, int k0, int n0, int lane) {
  int q   = (k0 >> 1) + (lane >> 4);
  int col = n0 + (lane & 15);
  return *(const v2f*)(base + q * ncol2 + col * 2);
}
// B fragment from row-major matrix (nodes): two strided b32 (compiler -> ds_load_2addr)
__device__ __forceinline__ v2f ldsB(const float* base, int ld, int k0, int n0, int lane) {
  int col = n0 + (lane & 15);
  int kk  = k0 + ((lane >> 4) << 1);
  v2f b; b.x = base[kk * ld + col]; b.y = base[(kk + 1) * ld + col]; return b;
}
// C/D (16x16 f32): VGPR e holds row m0+e (lanes 0-15) / m0+8+e (lanes 16-31)
__device__ __forceinline__ void stD(float* base, int ld, int m0, int n0, int lane, v8f d) {
  int col = n0 + (lane & 15);
  int r0  = m0 + ((lane >> 4) << 3);
#pragma unroll
  for (int e = 0; e < 8; ++e) base[(r0 + e) * ld + col] = d[e];
}
__device__ __forceinline__ v8f ldD(const float* base, int ld, int m0, int n0, int lane) {
  int col = n0 + (lane & 15);
  int r0  = m0 + ((lane >> 4) << 3);
  v8f v;
#pragma unroll
  for (int e = 0; e < 8; ++e) v[e] = base[(r0 + e) * ld + col];
  return v;
}
__device__ __forceinline__ float sigm(float x)  { return 1.0f / (1.0f + __expf(-x)); }
__device__ __forceinline__ float tanh1(float x) { return 1.0f - 2.0f / (__expf(2.0f * x) + 1.0f); }

__global__ __launch_bounds__(NT) void ggnn_kernel(
    const float* __restrict__ xg,   const float* __restrict__ in_mat,
    const float* __restrict__ out_mat, const float* __restrict__ maskG,
    const float* __restrict__ gW3w, const float* __restrict__ gb3w,
    const float* __restrict__ gW3u, const float* __restrict__ gb3u,
    const float* __restrict__ gW4w, const float* __restrict__ gb4w,
    const float* __restrict__ gW5w, const float* __restrict__ gb5w,
    const float* __restrict__ gW5u, const float* __restrict__ gb5u,
    const float* __restrict__ gWo,  const float* __restrict__ gbo,
    const float* __restrict__ gWrx, const float* __restrict__ gbrx,
    const float* __restrict__ gWry, const float* __restrict__ gbry,
    const float* __restrict__ gWr2, const float* __restrict__ gbr2,
    float* __restrict__ outp)
{
  extern __shared__ float sm[];
  const int tid  = threadIdx.x;
  const int lane = tid & 31;
  const int wave = tid >> 5;

  // ---- stage weights to LDS in pair-interleaved B-fragment layout ----
  // 32-col (H=32) matrices: i = q*64 + n*2 + j ; k = 2q+j
  for (int i = tid; i < 1024; i += NT) {
    int q = i >> 6, n = (i & 63) >> 1, j = i & 1, k = 2 * q + j;
    sm[OFF_W3UT + i] = gW3u[n * 32 + k];
    sm[OFF_W5UT + i] = gW5u[n * 32 + k];
  }
  for (int i = tid; i < 2048; i += NT) {
    int q = i >> 6, n = (i & 63) >> 1, j = i & 1, k = 2 * q + j;  // k in 0..63
    sm[OFF_W3WT + i] = gW3w[n * 64 + k];
    sm[OFF_W4WT + i] = gW4w[n * 64 + k];
    sm[OFF_W5WT + i] = gW5w[n * 64 + k];
  }
  // 16-col matrices: i = q*32 + n*2 + j
  for (int i = tid; i < 1024; i += NT) {
    int q = i >> 5, n = (i & 31) >> 1, j = i & 1, k = 2 * q + j;  // k in 0..63
    sm[OFF_WOT + i] = gWo[n * 64 + k];
  }
  for (int i = tid; i < 512; i += NT) {
    int q = i >> 5, n = (i & 31) >> 1, j = i & 1, k = 2 * q + j;  // k in 0..31
    sm[OFF_WRXT + i] = gWrx[n * 32 + k];
    sm[OFF_WRYT + i] = gWry[n * 32 + k];
  }
  for (int i = tid; i < 1280; i += NT) sm[OFF_MAT  + i] = in_mat[(i / 80) * 96 + 16 + (i % 80)];
  for (int i = tid; i < 1280; i += NT) sm[OFF_MATT + i] = out_mat[(16 + (i >> 4)) * 96 + (i & 15)];
  if (tid < 32) {
    sm[OFF_BIAS +       tid] = gb3w[tid];
    sm[OFF_BIAS +  32 + tid] = gb3u[tid];
    sm[OFF_BIAS +  64 + tid] = gb4w[tid];
    sm[OFF_BIAS +  96 + tid] = gb5w[tid];
    sm[OFF_BIAS + 128 + tid] = gb5u[tid];
  }
  if (tid < 16) {
    sm[OFF_BIAS + 160 + tid] = gbo[tid];
    sm[OFF_BIAS + 176 + tid] = gbrx[tid];
    sm[OFF_BIAS + 192 + tid] = gbry[tid];
    sm[OFF_BIAS + 208 + tid] = gWr2[tid];
  }
  if (tid == 0) sm[OFF_BIAS + 224] = gbr2[0];
  __syncthreads();

  float* nodes = sm + OFF_WAVE + wave * WAVE_FL;  // 96x32
  float* aBuf  = nodes + 3072;                    // 96x32 (later 96x16 readout)
  float* rnB   = aBuf + 3072;                     // 16x32 (later 16x16 rfcx)
  const float* matS  = sm + OFF_MAT;
  const float* matTS = sm + OFF_MATT;
  const float* bias  = sm + OFF_BIAS;
  const int b = blockIdx.x * WAVES + wave;
  const float* xb = xg + (size_t)b * 3072;

  // ---- init nodes = x0 ----
  for (int i = lane * 4; i < 3072; i += 128)
    *(v4f*)(nodes + i) = *(const v4f*)(xb + i);

  // ---- T=5 propagation steps ----
  for (int t = 0; t < 5; ++t) {
    // Phase A: a_in = mat(16x80) @ nodes[16:96];  a_out = mat.T(80x16) @ nodes[0:16]
    {
      v2f am[20];
#pragma unroll
      for (int ks = 0; ks < 20; ++ks) am[ks] = ldsA(matS, 80, 0, ks * 4, lane);
#pragma unroll
      for (int n = 0; n < 2; ++n) {
        v8f acc = {};
#pragma unroll
        for (int ks = 0; ks < 20; ++ks) {
          v2f bb = ldsB(nodes + 16 * 32, 32, ks * 4, n * 16, lane);
          acc = wmma4(am[ks], bb, acc);
        }
        stD(aBuf, 32, 0, n * 16, lane, acc);
      }
    }
    for (int m = 1; m < 6; ++m) {
      v2f am[4];
#pragma unroll
      for (int ks = 0; ks < 4; ++ks) am[ks] = ldsA(matTS, 16, (m - 1) * 16, ks * 4, lane);
#pragma unroll
      for (int n = 0; n < 2; ++n) {
        v8f acc = {};
#pragma unroll
        for (int ks = 0; ks < 4; ++ks) {
          v2f bb = ldsB(nodes, 32, ks * 4, n * 16, lane);
          acc = wmma4(am[ks], bb, acc);
        }
        stD(aBuf, 32, m * 16, n * 16, lane, acc);
      }
    }
    // Phase B: GRU gates per 16-row block; row block 0 uses W*w[:, :32], others W*w[:, 32:]
    for (int m = 0; m < 6; ++m) {
      const int kb = (m == 0) ? 0 : 32;
      v8f zk[2], hk[2], nd[2];
      v2f aN[8], aA[8];
#pragma unroll
      for (int ks = 0; ks < 8; ++ks) {
        aN[ks] = ldsA(nodes, 32, m * 16, ks * 4, lane);
        aA[ks] = ldsA(aBuf,  32, m * 16, ks * 4, lane);
      }
#pragma unroll
      for (int n = 0; n < 2; ++n) {
        v8f au = {}, az = {}, ar = {}, ah = {};
#pragma unroll
        for (int ks = 0; ks < 8; ++ks) {
          v2f bU = ldsBP(sm + OFF_W3UT, 64, ks * 4, n * 16, lane);
          v2f bZ = ldsBP(sm + OFF_W3WT, 64, kb + ks * 4, n * 16, lane);
          v2f bR = ldsBP(sm + OFF_W4WT, 64, kb + ks * 4, n * 16, lane);
          v2f bH = ldsBP(sm + OFF_W5WT, 64, kb + ks * 4, n * 16, lane);
          au = wmma4(aN[ks], bU, au);
          az = wmma4(aA[ks], bZ, az);
          ar = wmma4(aA[ks], bR, ar);
          ah = wmma4(aA[ks], bH, ah);
        }
        nd[n] = ldD(nodes, 32, m * 16, n * 16, lane);
        const int col = n * 16 + (lane & 15);
        const float bz = bias[col]      + bias[32 + col];   // b3w + b3u
        const float br = bias[64 + col] + bias[32 + col];   // b4w + b3u
        v8f z, rn;
#pragma unroll
        for (int e = 0; e < 8; ++e) {
          float u = au[e];
          z[e]  = sigm(az[e] + u + bz);
          float r = sigm(ar[e] + u + br);
          rn[e] = r * nd[n][e];
        }
        zk[n] = z; hk[n] = ah;
        stD(rnB, 32, 0, n * 16, lane, rn);  // stage (r*nodes) row-major for W5u GEMM
      }
      {
        v2f aR[8];
#pragma unroll
        for (int ks = 0; ks < 8; ++ks) aR[ks] = ldsA(rnB, 32, 0, ks * 4, lane);
#pragma unroll
        for (int n = 0; n < 2; ++n) {
          v8f acc = hk[n];
#pragma unroll
          for (int ks = 0; ks < 8; ++ks) {
            v2f bW = ldsBP(sm + OFF_W5UT, 64, ks * 4, n * 16, lane);
            acc = wmma4(aR[ks], bW, acc);
          }
          const int col = n * 16 + (lane & 15);
          const float bh = bias[96 + col] + bias[128 + col];  // b5w + b5u
          v8f nn;
#pragma unroll
          for (int e = 0; e < 8; ++e) {
            float h = tanh1(acc[e] + bh);
            nn[e]   = nd[n][e] + zk[n][e] * (h - nd[n][e]);
          }
          stD(nodes, 32, m * 16, n * 16, lane, nn);
        }
      }
    }
  }

  // ---- readout: out = tanh([nodes | x0] @ Wo.T + bo) -> aBuf (96x16) ----
  {
    const float bO = bias[160 + (lane & 15)];
    for (int m = 0; m < 6; ++m) {
      v8f acc = {};
#pragma unroll
      for (int ks = 0; ks < 8; ++ks) {
        v2f aN = ldsA(nodes, 32, m * 16, ks * 4, lane);
        v2f bW = ldsBP(sm + OFF_WOT, 32, ks * 4, 0, lane);
        acc = wmma4(aN, bW, acc);
      }
#pragma unroll
      for (int ks = 0; ks < 8; ++ks) {
        v2f aX = glbA(xb, 32, m * 16, ks * 4, lane);     // x0 streamed from global
        v2f bW = ldsBP(sm + OFF_WOT, 32, 32 + ks * 4, 0, lane);
        acc = wmma4(aX, bW, acc);
      }
      v8f o;
#pragma unroll
      for (int e = 0; e < 8; ++e) o[e] = tanh1(acc[e] + bO);
      stD(aBuf, 16, m * 16, 0, lane, o);
    }
  }
  // rfcx = tanh(nodes[0:16] @ Wrx.T + brx) -> rnB (16x16)
  {
    const float bX = bias[176 + (lane & 15)];
    v8f acc = {};
#pragma unroll
    for (int ks = 0; ks < 8; ++ks) {
      v2f aN = ldsA(nodes, 32, 0, ks * 4, lane);
      v2f bW = ldsBP(sm + OFF_WRXT, 32, ks * 4, 0, lane);
      acc = wmma4(aN, bW, acc);
    }
    v8f o;
#pragma unroll
    for (int e = 0; e < 8; ++e) o[e] = tanh1(acc[e] + bX);
    stD(rnB, 16, 0, 0, lane, o);
  }
  // rfcy = tanh(nodes[16:96] @ Wry.T + bry) -> compact into nodes[0..1280) as 80x16
  // (tile j writes floats [256j,256j+256) == rows <=8j+7, reads rows 16+16j.. : no overlap)
  {
    const float bY = bias[192 + (lane & 15)];
    for (int j = 0; j < 5; ++j) {
      v8f acc = {};
#pragma unroll
      for (int ks = 0; ks < 8; ++ks) {
        v2f aN = ldsA(nodes, 32, 16 + j * 16, ks * 4, lane);
        v2f bW = ldsBP(sm + OFF_WRYT, 32, ks * 4, 0, lane);
        acc = wmma4(aN, bW, acc);
      }
      v8f o;
#pragma unroll
      for (int e = 0; e < 8; ++e) o[e] = tanh1(acc[e] + bY);
      stD(nodes + j * 256, 16, 0, 0, lane, o);
    }
  }
  // ---- final assembly & non-temporal global stores (write-once 340MB output) ----
  {
    float* outBase = outp + (size_t)b * 20736;  // (16,81,16)
    {   // routput: final[b,c,0,:] = out[c,:]
      int c = lane >> 1, hh = (lane & 1) * 8;
      const float* src = aBuf + c * 16 + hh;
      v4f v0 = *(const v4f*)(src);
      v4f v1 = *(const v4f*)(src + 4);
      __builtin_nontemporal_store(v0, (v4f*)(outBase + c * 1296 + hh));
      __builtin_nontemporal_store(v1, (v4f*)(outBase + c * 1296 + hh + 4));
    }
    float w2[16];
#pragma unroll
    for (int k = 0; k < 16; ++k) w2[k] = bias[208 + k];
    const float bR2 = bias[224];
    for (int p = lane; p < 1280; p += 32) {
      int c = p / 80, o = p - c * 80;
      const float* fx = rnB   + c * 16;
      const float* fy = nodes + o * 16;
      float dot = 0.f;
#pragma unroll
      for (int k = 0; k < 16; ++k) dot += fx[k] * fy[k] * w2[k];
      float s = sigm(dot + bR2) * maskG[p];           // mask[c*80+o] == mask[p]
      const float* oo = aBuf + (16 + o) * 16;          // ooutput row
      float* dst = outBase + c * 1296 + (1 + o) * 16;  // final[b,c,1+o,:]
#pragma unroll
      for (int q = 0; q < 4; ++q) {
        v4f v = *(const v4f*)(oo + q * 4);
        v *= s;
        __builtin_nontemporal_store(v, (v4f*)(dst + q * 4));
      }
    }
  }
}

extern "C" void kernel_launch(void* const* d_in, const int* in_sizes, int n_in,
                              void* d_out, int out_size, void* d_ws, size_t ws_size,
                              hipStream_t stream) {
  (void)in_sizes; (void)n_in; (void)d_ws; (void)ws_size; (void)out_size;
  const float* x    = (const float*)d_in[0];
  const float* inm  = (const float*)d_in[1];
  const float* outm = (const float*)d_in[2];
  const float* mask = (const float*)d_in[3];
  const float* W3w  = (const float*)d_in[4];
  const float* b3w  = (const float*)d_in[5];
  const float* W3u  = (const float*)d_in[6];
  const float* b3u  = (const float*)d_in[7];
  const float* W4w  = (const float*)d_in[8];
  const float* b4w  = (const float*)d_in[9];
  const float* W5w  = (const float*)d_in[10];
  const float* b5w  = (const float*)d_in[11];
  const float* W5u  = (const float*)d_in[12];
  const float* b5u  = (const float*)d_in[13];
  const float* Wo   = (const float*)d_in[14];
  const float* bo   = (const float*)d_in[15];
  const float* Wrx  = (const float*)d_in[16];
  const float* brx  = (const float*)d_in[17];
  const float* Wry  = (const float*)d_in[18];
  const float* bry  = (const float*)d_in[19];
  const float* Wr2  = (const float*)d_in[20];
  const float* br2  = (const float*)d_in[21];

  hipFuncSetAttribute((const void*)ggnn_kernel,
                      hipFuncAttributeMaxDynamicSharedMemorySize, SMEM_BYTES);

  ggnn_kernel<<<dim3(4096 / WAVES), dim3(NT), SMEM_BYTES, stream>>>(
      x, inm, outm, mask, W3w, b3w, W3u, b3u, W4w, b4w, W5w, b5w, W5u, b5u,
      Wo, bo, Wrx, brx, Wry, bry, Wr2, br2, (float*)d_out);
}